// Net_2181843386916
// MI455X (gfx1250) — compile-verified
//
#include <hip/hip_runtime.h>
#include <hip/hip_bf16.h>

typedef __attribute__((ext_vector_type(2))) float v2f;
typedef __attribute__((ext_vector_type(8))) float v8f;

#define NEG_SLOPE 0.2f
#define EPS_F 1e-16f

// ---------------------------------------------------------------------------
// Native CDNA5 float atomics (no-return forms -> STOREcnt; s_endpgm does an
// implicit wait-idle, so completion before kernel end is guaranteed).
// Inline asm guarantees GLOBAL_ATOMIC_ADD_F32 / GLOBAL_ATOMIC_MAX_NUM_F32
// instead of a possible CAS-loop expansion of the generic HIP atomics.
// ---------------------------------------------------------------------------
__device__ __forceinline__ void atomicAddF32(float* p, float v) {
#if defined(__gfx1250__)
    asm volatile("global_atomic_add_f32 %0, %1, off" : : "v"(p), "v"(v) : "memory");
#else
    atomicAdd(p, v);
#endif
}

__device__ __forceinline__ void atomicMaxF32(float* p, float v) {
#if defined(__gfx1250__)
    asm volatile("global_atomic_max_num_f32 %0, %1, off" : : "v"(p), "v"(v) : "memory");
#else
    if (v >= 0.0f) atomicMax((int*)p, __float_as_int(v));
    else           atomicMin((unsigned int*)p, __float_as_uint(v));
#endif
}

// ---------------------------------------------------------------------------
// Utility: fill a buffer with a 32-bit pattern (used for -inf / 0 init).
// ---------------------------------------------------------------------------
__global__ __launch_bounds__(256) void fill_u32(unsigned int* __restrict__ p,
                                                unsigned int val, long n) {
    long i = (long)blockIdx.x * blockDim.x + threadIdx.x;
    if (i < n) p[i] = val;
}

// ---------------------------------------------------------------------------
// GEMM  H[N,128] = X[N,128] @ W[128,128]  using V_WMMA_F32_16X16X4_F32.
// Block = 256 threads = 8 waves; each wave computes one 16-row tile across
// all 128 columns (8 col-tiles). W is staged in LDS (64 KB of the 320 KB WGP
// pool); A is preloaded into 32 v2f registers per lane.
//
// Per ISA layouts (wave32):
//   A 16x4 f32 : lane L -> M = L&15, k = vgpr + 2*(L>>4)
//   B  4x16 f32: lane L -> N = L&15, k = vgpr + 2*(L>>4)
//   C/D 16x16  : vgpr v -> M = v + 8*(L>>4), N = L&15
// ---------------------------------------------------------------------------
__global__ __launch_bounds__(256) void gemm_wmma(const float* __restrict__ X,
                                                 const float* __restrict__ W,
                                                 float* __restrict__ H,
                                                 int n_row_tiles, long nrows) {
    __shared__ float Wlds[128 * 128];

    const int tid = threadIdx.x;
    // Cooperative load of W into LDS: 4096 float4 / 256 threads = 16 each.
    {
        const float4* Wg = (const float4*)W;
        float4* Wl = (float4*)Wlds;
#pragma unroll
        for (int i = 0; i < 16; ++i) {
            int idx = i * 256 + tid;
            Wl[idx] = Wg[idx];
        }
    }
    __syncthreads();

    const int wave = tid >> 5;
    const int lane = tid & 31;
    const int tile = blockIdx.x * 8 + wave;
    if (tile >= n_row_tiles) return;

    const int m     = lane & 15;        // row-in-tile (A) / col-in-tile (B,C)
    const int khalf = (lane >> 4) * 2;  // 0 or 2

    long row = (long)tile * 16 + m;
    if (row >= nrows) row = nrows - 1;  // safe clamp (N%16==0 in practice)
    const float* xrow = X + row * 128;

    // Preload A: k = 4t + khalf + {0,1}
    v2f a[32];
#pragma unroll
    for (int t = 0; t < 32; ++t) {
        const int k0 = 4 * t + khalf;
        a[t].x = xrow[k0];
        a[t].y = xrow[k0 + 1];
    }

    const long rbase = (long)tile * 16 + (lane >> 4) * 8;

#pragma unroll 1
    for (int ct = 0; ct < 8; ++ct) {
        const int col = ct * 16 + m;
        v8f acc = {};
#pragma unroll
        for (int t = 0; t < 32; ++t) {
            const int k0 = 4 * t + khalf;
            v2f b;
            b.x = Wlds[k0 * 128 + col];
            b.y = Wlds[(k0 + 1) * 128 + col];
            acc = __builtin_amdgcn_wmma_f32_16x16x4_f32(
                false, a[t], false, b, (short)0, acc, false, false);
        }
#pragma unroll
        for (int v = 0; v < 8; ++v) {
            const long r = rbase + v;
            if (r < nrows) H[r * 128 + col] = acc[v];
        }
    }
}

// ---------------------------------------------------------------------------
// Per-node attention scores: a_s[n] = h[n,:]·att_src, a_d[n] = h[n,:]·att_dst
// One wave per node; float4 per lane; shfl_xor tree reduction (wave32).
// ---------------------------------------------------------------------------
__global__ __launch_bounds__(256) void attn_scores(const float* __restrict__ H,
                                                   const float* __restrict__ att_s,
                                                   const float* __restrict__ att_d,
                                                   float* __restrict__ a_s,
                                                   float* __restrict__ a_d, int n) {
    const int node = blockIdx.x * 8 + (threadIdx.x >> 5);
    const int lane = threadIdx.x & 31;
    if (node >= n) return;
    const float4 hv = ((const float4*)(H + (long)node * 128))[lane];
    const float4 sv = ((const float4*)att_s)[lane];
    const float4 dv = ((const float4*)att_d)[lane];
    float s = hv.x * sv.x + hv.y * sv.y + hv.z * sv.z + hv.w * sv.w;
    float d = hv.x * dv.x + hv.y * dv.y + hv.z * dv.z + hv.w * dv.w;
#pragma unroll
    for (int off = 16; off; off >>= 1) {
        s += __shfl_xor(s, off, 32);
        d += __shfl_xor(d, off, 32);
    }
    if (lane == 0) { a_s[node] = s; a_d[node] = d; }
}

// ---------------------------------------------------------------------------
// Edge pass 1: e = leaky_relu(a_s[src]+a_d[dst]); segment max into m[dst]
// via native GLOBAL_ATOMIC_MAX_NUM_F32 (m initialized to -inf bits).
// Edges [0,E) from edge_index, [E,E+N) are self-loops.
// ---------------------------------------------------------------------------
__global__ __launch_bounds__(256) void edge_logits_max(const int* __restrict__ ei,
                                                       const float* __restrict__ a_s,
                                                       const float* __restrict__ a_d,
                                                       float* __restrict__ evalb,
                                                       float* __restrict__ m,
                                                       int E, int N) {
    const int e = blockIdx.x * blockDim.x + threadIdx.x;
    if (e >= E + N) return;
    const int src = (e < E) ? ei[e]     : (e - E);
    const int dst = (e < E) ? ei[E + e] : (e - E);
    float v = a_s[src] + a_d[dst];
    v = (v > 0.0f) ? v : NEG_SLOPE * v;
    evalb[e] = v;
    atomicMaxF32(m + dst, v);
}

// Edge pass 2: ex = exp(e - m[dst]); segment sum into denom[dst].
__global__ __launch_bounds__(256) void edge_expsum(const int* __restrict__ ei,
                                                   const float* __restrict__ evalb,
                                                   const float* __restrict__ m,
                                                   float* __restrict__ denom,
                                                   float* __restrict__ exb,
                                                   int E, int N) {
    const int e = blockIdx.x * blockDim.x + threadIdx.x;
    if (e >= E + N) return;
    const int dst = (e < E) ? ei[E + e] : (e - E);
    const float ex = __expf(evalb[e] - m[dst]);
    exb[e] = ex;
    atomicAddF32(denom + dst, ex);
}

// Edge pass 3: agg[dst,:] += alpha * h[src,:].  One wave per edge: float4
// gather per lane + 4 native GLOBAL_ATOMIC_ADD_F32 per lane.
__global__ __launch_bounds__(256) void edge_aggregate(const int* __restrict__ ei,
                                                      const float* __restrict__ exb,
                                                      const float* __restrict__ denom,
                                                      const float* __restrict__ H,
                                                      float* __restrict__ agg,
                                                      int E, int N) {
    const long tid = (long)blockIdx.x * blockDim.x + threadIdx.x;
    const long e = tid >> 5;
    const int q = (int)(tid & 31);
    if (e >= (long)E + N) return;
    const int src = (e < E) ? ei[e]     : (int)(e - E);
    const int dst = (e < E) ? ei[E + e] : (int)(e - E);
    const float alpha = exb[e] / (denom[dst] + EPS_F);
    const float4 hv = ((const float4*)(H + (long)src * 128))[q];
    float* out = agg + (long)dst * 128 + q * 4;
    atomicAddF32(out + 0, alpha * hv.x);
    atomicAddF32(out + 1, alpha * hv.y);
    atomicAddF32(out + 2, alpha * hv.z);
    atomicAddF32(out + 3, alpha * hv.w);
}

// Fused bias (+ optional ReLU), in place.
__global__ __launch_bounds__(256) void bias_act(float* __restrict__ buf,
                                                const float* __restrict__ bias,
                                                long total, int relu) {
    const long i = (long)blockIdx.x * blockDim.x + threadIdx.x;
    if (i >= total) return;
    float v = buf[i] + bias[i & 127];
    if (relu) v = (v > 0.0f) ? v : 0.0f;
    buf[i] = v;
}

// Link prediction: pred[p] = x[first[p],:] · x[second[p],:].  Wave per pair.
__global__ __launch_bounds__(256) void pred_dot(const float* __restrict__ X,
                                                const int* __restrict__ eli,
                                                float* __restrict__ out, int P) {
    const int p = blockIdx.x * 8 + (threadIdx.x >> 5);
    const int lane = threadIdx.x & 31;
    if (p >= P) return;
    const int i = eli[p];
    const int j = eli[P + p];
    const float4 u = ((const float4*)(X + (long)i * 128))[lane];
    const float4 v = ((const float4*)(X + (long)j * 128))[lane];
    float s = u.x * v.x + u.y * v.y + u.z * v.z + u.w * v.w;
#pragma unroll
    for (int off = 16; off; off >>= 1) s += __shfl_xor(s, off, 32);
    if (lane == 0) out[p] = s;
}

// ---------------------------------------------------------------------------
// Host launch
// ---------------------------------------------------------------------------
extern "C" void kernel_launch(void* const* d_in, const int* in_sizes, int n_in,
                              void* d_out, int out_size, void* d_ws, size_t ws_size,
                              hipStream_t stream) {
    const float* x0  = (const float*)d_in[0];
    const int*   ei  = (const int*)d_in[1];
    const int*   eli = (const int*)d_in[2];
    const float* W1  = (const float*)d_in[3];
    const float* as1 = (const float*)d_in[4];
    const float* ad1 = (const float*)d_in[5];
    const float* b1  = (const float*)d_in[6];
    const float* W2  = (const float*)d_in[7];
    const float* as2 = (const float*)d_in[8];
    const float* ad2 = (const float*)d_in[9];
    const float* b2  = (const float*)d_in[10];

    const int N = in_sizes[0] / 128;
    const int E = in_sizes[1] / 2;
    const int P = in_sizes[2] / 2;
    const int Etot = E + N;
    const long NC = (long)N * 128;

    // Workspace layout (floats): two big ping-pong buffers + small vectors.
    float* base  = (float*)d_ws;
    size_t off   = 0;
    float* big0  = base + off; off += (size_t)NC;   // h (layer GEMM output)
    float* big1  = base + off; off += (size_t)NC;   // agg / x
    float* a_s   = base + off; off += (size_t)N;
    float* a_d   = base + off; off += (size_t)N;
    float* mbuf  = base + off; off += (size_t)N;
    float* dbuf  = base + off; off += (size_t)N;
    float* evalb = base + off; off += (size_t)Etot;
    float* exb   = base + off; off += (size_t)Etot;
    (void)ws_size; (void)n_in; (void)out_size;

    const int ntiles     = (N + 15) / 16;
    const int gemmBlocks = (ntiles + 7) / 8;
    const int nodeWaves  = (N + 7) / 8;
    const int edgeBlocks = (Etot + 255) / 256;
    const int aggBlocks  = (int)(((long)Etot * 32 + 255) / 256);
    const int ncBlocks   = (int)((NC + 255) / 256);
    const int nBlocks    = (N + 255) / 256;

    auto gat_layer = [&](const float* xin, const float* W, const float* atts,
                         const float* attd, const float* bias,
                         float* hbuf, float* aggbuf, int relu) {
        gemm_wmma<<<gemmBlocks, 256, 0, stream>>>(xin, W, hbuf, ntiles, (long)N);
        attn_scores<<<nodeWaves, 256, 0, stream>>>(hbuf, atts, attd, a_s, a_d, N);
        fill_u32<<<nBlocks, 256, 0, stream>>>((unsigned int*)mbuf, 0xFF800000u, N); // -inf
        fill_u32<<<nBlocks, 256, 0, stream>>>((unsigned int*)dbuf, 0u, N);
        fill_u32<<<ncBlocks, 256, 0, stream>>>((unsigned int*)aggbuf, 0u, NC);
        edge_logits_max<<<edgeBlocks, 256, 0, stream>>>(ei, a_s, a_d, evalb, mbuf, E, N);
        edge_expsum<<<edgeBlocks, 256, 0, stream>>>(ei, evalb, mbuf, dbuf, exb, E, N);
        edge_aggregate<<<aggBlocks, 256, 0, stream>>>(ei, exb, dbuf, hbuf, aggbuf, E, N);
        bias_act<<<ncBlocks, 256, 0, stream>>>(aggbuf, bias, NC, relu);
    };

    // Layer 1: x0 -> h1(big0) -> agg/x1(big1), ReLU.
    gat_layer(x0, W1, as1, ad1, b1, big0, big1, 1);
    // Layer 2: x1(big1) -> h2(big0) -> agg/x2(big1), no ReLU.
    // (big1 is free for agg once the layer-2 GEMM has consumed it; stream order
    //  guarantees that.)
    gat_layer(big1, W2, as2, ad2, b2, big0, big1, 0);

    // Link prediction dots.
    pred_dot<<<(P + 7) / 8, 256, 0, stream>>>(big1, eli, (float*)d_out, P);
}